// spConvnet_subM_41223096107609
// MI455X (gfx1250) — compile-verified
//
#include <hip/hip_runtime.h>
#include <math.h>

typedef float v2f __attribute__((ext_vector_type(2)));
typedef float v8f __attribute__((ext_vector_type(8)));

#define DGRID 120
#define DG3   (120*120*120)        // 1,728,000 per batch
#define S3 58
#define S3C (58*58*58)             // 195,112
#define S4 29
#define S4C (29*29*29)             // 24,389
#define S5 14
#define S5C (14*14*14)             // 2,744
#define S6 12
#define S6C (12*12*12)             // 1,728

// ---------------- fills ----------------
__global__ void fillF_k(float* p, float v, size_t n) {
  for (size_t i = (size_t)blockIdx.x * blockDim.x + threadIdx.x; i < n;
       i += (size_t)gridDim.x * blockDim.x) p[i] = v;
}
__global__ void fillI_k(int* p, int v, size_t n) {
  for (size_t i = (size_t)blockIdx.x * blockDim.x + threadIdx.x; i < n;
       i += (size_t)gridDim.x * blockDim.x) p[i] = v;
}

// ---------------- scatter points into dense grids ----------------
__global__ void scatter_k(const float* feats, const int* indices,
                          float* featG, int* idxG, int N) {
  int i = blockIdx.x * blockDim.x + threadIdx.x;
  if (i >= N) return;
  int b = indices[i*4+0], z = indices[i*4+1], y = indices[i*4+2], x = indices[i*4+3];
  int flat = ((b*DGRID + z)*DGRID + y)*DGRID + x;
  featG[flat] = feats[i];
  idxG[flat]  = i;
}

// ---------------- conv1: 7x7x7, 1->32, gather at active points ----------------
__global__ __launch_bounds__(32) void conv1_k(const float* featG, const int* indices,
                                              const float* W1, const float* b1,
                                              float* y1, int N) {
  int p = blockIdx.x;
  if (p >= N) return;
  int c = threadIdx.x;  // 0..31
  int b = indices[p*4+0], z = indices[p*4+1], y = indices[p*4+2], x = indices[p*4+3];
  float acc = 0.f;
  for (int t = 0; t < 343; ++t) {
    int kz = t / 49, ky = (t / 7) % 7, kx = t % 7;
    int zz = z + kz - 3, yy = y + ky - 3, xx = x + kx - 3;
    if ((unsigned)zz < (unsigned)DGRID && (unsigned)yy < (unsigned)DGRID &&
        (unsigned)xx < (unsigned)DGRID) {
      float f = featG[((b*DGRID + zz)*DGRID + yy)*DGRID + xx];
      if (f != 0.f) acc = fmaf(f, W1[t*32 + c], acc);
    }
  }
  float v = acc + b1[c];
  y1[p*32 + c] = v > 0.f ? v : 0.f;
}

// ---------------- conv2: 5x5x5, 32->40, WMMA f32 16x16x4 over im2col taps ----
// Block = 96 threads = 3 waves; each wave owns one N-tile of 16 output channels.
__global__ __launch_bounds__(96) void conv2_k(const float* y1, const int* idxG,
                                              const int* indices, const float* W2,
                                              const float* b2, float* y2, int N) {
  __shared__ float Al[16][33];   // A tile: 16 points x 32 in-channels (padded stride)
  __shared__ float Bl[32][48];   // B tile: 32 in-ch x 48 out-ch (40 real, zero pad)
  __shared__ int   pc[16][4];    // per-point (b,z,y,x); b=-1 if invalid

  const int tile = blockIdx.x;
  const int tid  = threadIdx.x;
  const int lane = tid & 31;
  const int wave = tid >> 5;          // 0..2 -> N-tile
  const int hi   = lane >> 4;         // lane half selects K pair
  const int Mrow = lane & 15;
  const int ncol = wave * 16 + Mrow;  // output channel (0..47)

  if (tid < 16) {
    int p = tile * 16 + tid;
    if (p < N) {
      pc[tid][0] = indices[p*4+0]; pc[tid][1] = indices[p*4+1];
      pc[tid][2] = indices[p*4+2]; pc[tid][3] = indices[p*4+3];
    } else {
      pc[tid][0] = -1; pc[tid][1] = 0; pc[tid][2] = 0; pc[tid][3] = 0;
    }
  }
  __syncthreads();

  v8f acc = {};
  for (int t = 0; t < 125; ++t) {
    int kz = t / 25, ky = (t / 5) % 5, kx = t % 5;
    // stage A: gather neighbor features (0 when no active neighbor)
    for (int s = tid; s < 512; s += 96) {
      int i = s >> 5, c = s & 31;
      float v = 0.f;
      int bb = pc[i][0];
      if (bb >= 0) {
        int zz = pc[i][1] + kz - 2, yy = pc[i][2] + ky - 2, xx = pc[i][3] + kx - 2;
        if ((unsigned)zz < (unsigned)DGRID && (unsigned)yy < (unsigned)DGRID &&
            (unsigned)xx < (unsigned)DGRID) {
          int q = idxG[((bb*DGRID + zz)*DGRID + yy)*DGRID + xx];
          if (q >= 0) v = y1[q*32 + c];
        }
      }
      Al[i][c] = v;
    }
    // stage B: tap weight slice W2[t][k][n]
    for (int s = tid; s < 32*48; s += 96) {
      int k = s / 48, n = s % 48;
      Bl[k][n] = (n < 40) ? W2[(t*32 + k)*40 + n] : 0.f;
    }
    if (t + 1 < 125) __builtin_prefetch(&W2[(t + 1) * 32 * 40], 0, 1);
    __syncthreads();

    // 8 WMMA f32 16x16x4 steps cover K=32
    for (int k0 = 0; k0 < 32; k0 += 4) {
      v2f a, bv;
      a.x  = Al[Mrow][k0 + 2*hi];
      a.y  = Al[Mrow][k0 + 2*hi + 1];
      bv.x = Bl[k0 + 2*hi][ncol];
      bv.y = Bl[k0 + 2*hi + 1][ncol];
      acc = __builtin_amdgcn_wmma_f32_16x16x4_f32(false, a, false, bv,
                                                  (short)0, acc, false, false);
    }
    __syncthreads();
  }

  // D layout: lane holds column ncol; rows M = r + 8*hi
  for (int r = 0; r < 8; ++r) {
    int m = r + 8*hi;
    int p = tile * 16 + m;
    if (p < N && ncol < 40) {
      float v = acc[r] + b2[ncol];
      y2[p*40 + ncol] = v > 0.f ? v : 0.f;
    }
  }
}

// ---------------- conv3: 6x6x6 stride2, 40->50, scatter with f32 atomics -----
__global__ __launch_bounds__(64) void conv3_k(const float* y2, const int* idxG,
                                              const int* indices, const float* W3,
                                              float* out3, float* m3, int N) {
  int p = blockIdx.x;
  int tid = threadIdx.x;
  __shared__ float ys[40];
  __shared__ int own;
  int b = indices[p*4+0], z = indices[p*4+1], y = indices[p*4+2], x = indices[p*4+3];
  if (tid == 0) own = (idxG[((b*DGRID + z)*DGRID + y)*DGRID + x] == p) ? 1 : 0;
  if (tid < 40) ys[tid] = y2[p*40 + tid];
  __syncthreads();
  if (!own) return;   // dedupe duplicate-coordinate points (uniform branch)

  int ozl[3], kzl[3], nz = 0;
  int oyl[3], kyl[3], ny = 0;
  int oxl[3], kxl[3], nx = 0;
  for (int k = 0; k < 6; ++k) {
    int tz = z - k; if (tz >= 0 && !(tz & 1) && (tz >> 1) <= S3-1) { ozl[nz] = tz >> 1; kzl[nz] = k; ++nz; }
    int ty = y - k; if (ty >= 0 && !(ty & 1) && (ty >> 1) <= S3-1) { oyl[ny] = ty >> 1; kyl[ny] = k; ++ny; }
    int tx = x - k; if (tx >= 0 && !(tx & 1) && (tx >> 1) <= S3-1) { oxl[nx] = tx >> 1; kxl[nx] = k; ++nx; }
  }
  for (int iz = 0; iz < nz; ++iz)
    for (int iy = 0; iy < ny; ++iy)
      for (int ix = 0; ix < nx; ++ix) {
        int t3 = (kzl[iz]*6 + kyl[iy])*6 + kxl[ix];
        int cell = ((b*S3 + ozl[iz])*S3 + oyl[iy])*S3 + oxl[ix];
        if (tid == 0) m3[cell] = 1.f;
        if (tid < 50) {
          const float* w = W3 + (t3*40)*50 + tid;
          float s = 0.f;
          for (int ic = 0; ic < 40; ++ic) s = fmaf(ys[ic], w[ic*50], s);
          atomicAdd(&out3[cell*50 + tid], s);
        }
      }
}

// ---------------- post conv3: bias + relu where mask active ----------------
__global__ void post3_k(float* out3, const float* m3, const float* b3) {
  int i = blockIdx.x * blockDim.x + threadIdx.x;
  if (i >= 2*S3C) return;
  if (m3[i] > 0.f) {
    int base = i * 50;
    for (int c = 0; c < 50; ++c) {
      float v = out3[base + c] + b3[c];
      out3[base + c] = v > 0.f ? v : 0.f;
    }
  }
}

// ---------------- maxpool 2,2 over 58^3 x50 -> 29^3 x50 ----------------
__global__ void pool4_k(const float* out3, float* x4) {
  int i = blockIdx.x * blockDim.x + threadIdx.x;
  if (i >= 2*S4C*50) return;
  int c = i % 50; int r = i / 50;
  int x = r % S4; r /= S4; int y = r % S4; r /= S4; int z = r % S4; int b = r / S4;
  float m = 0.f;  // reference reduce_window init = 0.0, inputs >= 0
  for (int dz = 0; dz < 2; ++dz)
    for (int dy = 0; dy < 2; ++dy)
      for (int dx = 0; dx < 2; ++dx) {
        float v = out3[(((b*S3 + 2*z+dz)*S3 + 2*y+dy)*S3 + 2*x+dx)*50 + c];
        m = v > m ? v : m;
      }
  x4[i] = m;
}

// ---------------- mask pools ----------------
__global__ void m29_k(const float* m3, float* m29) {
  int i = blockIdx.x * blockDim.x + threadIdx.x;
  if (i >= 2*S4C) return;
  int x = i % S4; int r = i / S4; int y = r % S4; r /= S4; int z = r % S4; int b = r / S4;
  float m = 0.f;
  for (int dz = 0; dz < 2; ++dz) for (int dy = 0; dy < 2; ++dy) for (int dx = 0; dx < 2; ++dx) {
    float v = m3[((b*S3 + 2*z+dz)*S3 + 2*y+dy)*S3 + 2*x+dx];
    m = v > m ? v : m;
  }
  m29[i] = m;
}
__global__ void m14_k(const float* m29, float* m14) {
  int i = blockIdx.x * blockDim.x + threadIdx.x;
  if (i >= 2*S5C) return;
  int x = i % S5; int r = i / S5; int y = r % S5; r /= S5; int z = r % S5; int b = r / S5;
  float m = 0.f;
  for (int dz = 0; dz < 3; ++dz) for (int dy = 0; dy < 3; ++dy) for (int dx = 0; dx < 3; ++dx) {
    float v = m29[((b*S4 + 2*z+dz)*S4 + 2*y+dy)*S4 + 2*x+dx];
    m = v > m ? v : m;
  }
  m14[i] = m;
}
__global__ void m12_k(const float* m14, float* m12) {
  int i = blockIdx.x * blockDim.x + threadIdx.x;
  if (i >= 2*S6C) return;
  int x = i % S6; int r = i / S6; int y = r % S6; r /= S6; int z = r % S6; int b = r / S6;
  float m = 0.f;
  for (int dz = 0; dz < 3; ++dz) for (int dy = 0; dy < 3; ++dy) for (int dx = 0; dx < 3; ++dx) {
    float v = m14[((b*S5 + z+dz)*S5 + y+dy)*S5 + x+dx];
    m = v > m ? v : m;
  }
  m12[i] = m;
}

// ---------------- conv4: 3^3 stride2, 50->30, dense ----------------
__global__ void conv4_k(const float* x4, const float* m14, const float* W4,
                        const float* b4, float* out4) {
  int i = blockIdx.x * blockDim.x + threadIdx.x;
  if (i >= 2*S5C*30) return;
  int oc = i % 30; int cell = i / 30;
  int x = cell % S5; int r = cell / S5; int y = r % S5; r /= S5; int z = r % S5; int b = r / S5;
  if (m14[cell] <= 0.f) { out4[i] = 0.f; return; }
  float acc = 0.f;
  for (int t = 0; t < 27; ++t) {
    int kz = t / 9, ky = (t / 3) % 3, kx = t % 3;
    const float* xp = x4 + (((b*S4 + 2*z+kz)*S4 + 2*y+ky)*S4 + 2*x+kx)*50;
    const float* wp = W4 + (t*50)*30 + oc;
    for (int ic = 0; ic < 50; ++ic) acc = fmaf(xp[ic], wp[ic*30], acc);
  }
  float v = acc + b4[oc];
  out4[i] = v > 0.f ? v : 0.f;
}

// ---------------- conv5: 3^3 stride1, 30->10, dense ----------------
__global__ void conv5_k(const float* out4, const float* m12, const float* W5,
                        const float* b5, float* out5) {
  int i = blockIdx.x * blockDim.x + threadIdx.x;
  if (i >= 2*S6C*10) return;
  int oc = i % 10; int cell = i / 10;
  int x = cell % S6; int r = cell / S6; int y = r % S6; r /= S6; int z = r % S6; int b = r / S6;
  if (m12[cell] <= 0.f) { out5[i] = 0.f; return; }
  float acc = 0.f;
  for (int t = 0; t < 27; ++t) {
    int kz = t / 9, ky = (t / 3) % 3, kx = t % 3;
    const float* xp = out4 + (((b*S5 + z+kz)*S5 + y+ky)*S5 + x+kx)*30;
    const float* wp = W5 + (t*30)*10 + oc;
    for (int ic = 0; ic < 30; ++ic) acc = fmaf(xp[ic], wp[ic*10], acc);
  }
  float v = acc + b5[oc];
  out5[i] = v > 0.f ? v : 0.f;
}

// ---------------- pool 2,2 (12->6) + channel-major flatten ----------------
__global__ void poolflat_k(const float* out5, float* flat) {
  int i = blockIdx.x * blockDim.x + threadIdx.x;
  if (i >= 2*10*216) return;
  int x = i % 6; int r = i / 6; int y = r % 6; r /= 6; int z = r % 6; r /= 6;
  int c = r % 10; int b = r / 10;
  float m = 0.f;
  for (int dz = 0; dz < 2; ++dz) for (int dy = 0; dy < 2; ++dy) for (int dx = 0; dx < 2; ++dx) {
    float v = out5[(((b*S6 + 2*z+dz)*S6 + 2*y+dy)*S6 + 2*x+dx)*10 + c];
    m = v > m ? v : m;
  }
  flat[b*2160 + c*216 + z*36 + y*6 + x] = m;
}

// ---------------- fc1: (2x2160)@(2160x500) via WMMA f32, 1 wave / N-tile ----
__global__ __launch_bounds__(32) void fc1_k(const float* flat, const float* w,
                                            const float* bias, float* a1) {
  int nt = blockIdx.x;                 // 32 tiles of 16 cols (500 real)
  int lane = threadIdx.x;
  int hi = lane >> 4;
  int M = lane & 15;
  int n = nt * 16 + M;
  bool nok = n < 500;
  v8f acc = {};
  for (int k0 = 0; k0 < 2160; k0 += 4) {
    int kk = k0 + 2*hi;
    v2f a, bv;
    a.x  = (M < 2) ? flat[M*2160 + kk]     : 0.f;
    a.y  = (M < 2) ? flat[M*2160 + kk + 1] : 0.f;
    bv.x = nok ? w[(long)n*2160 + kk]     : 0.f;
    bv.y = nok ? w[(long)n*2160 + kk + 1] : 0.f;
    acc = __builtin_amdgcn_wmma_f32_16x16x4_f32(false, a, false, bv,
                                                (short)0, acc, false, false);
  }
  if (lane < 16) {
    int nn = nt * 16 + lane;
    if (nn < 500) {
      float v0 = acc[0] + bias[nn];   // row M=0 (batch 0)
      float v1 = acc[1] + bias[nn];   // row M=1 (batch 1)
      a1[nn]        = v0 > 0.f ? v0 : 0.f;
      a1[500 + nn]  = v1 > 0.f ? v1 : 0.f;
    }
  }
}

// ---------------- fc2 / fc3 ----------------
__global__ void fc2_k(const float* a1, const float* w, const float* bias, float* a2) {
  int i = blockIdx.x * blockDim.x + threadIdx.x;
  if (i >= 2*200) return;
  int j = i % 200, b = i / 200;
  float s = bias[j];
  for (int k = 0; k < 500; ++k) s = fmaf(a1[b*500 + k], w[j*500 + k], s);
  a2[i] = s > 0.f ? s : 0.f;
}
__global__ void fc3_k(const float* a2, const float* w, const float* bias, float* a3) {
  int i = blockIdx.x * blockDim.x + threadIdx.x;
  if (i >= 2*50) return;
  int j = i % 50, b = i / 50;
  float s = bias[j];
  for (int k = 0; k < 200; ++k) s = fmaf(a2[b*200 + k], w[j*200 + k], s);
  a3[i] = tanhf(s);
}

// ---------------- fc4 + row-normalize ----------------
__global__ __launch_bounds__(64) void fc4norm_k(const float* a3, const float* w,
                                                const float* bias, float* out) {
  __shared__ float so[6];
  int tid = threadIdx.x;
  if (tid < 6) {
    int b = tid / 3, j = tid % 3;
    float s = bias[j];
    for (int k = 0; k < 50; ++k) s = fmaf(a3[b*50 + k], w[j*50 + k], s);
    so[tid] = s;
  }
  __syncthreads();
  if (tid < 6) {
    int b = tid / 3;
    float n = sqrtf(so[b*3]*so[b*3] + so[b*3+1]*so[b*3+1] + so[b*3+2]*so[b*3+2]);
    n = n > 1e-12f ? n : 1e-12f;
    out[tid] = so[tid] / n;
  }
}

// ---------------- host launcher ----------------
extern "C" void kernel_launch(void* const* d_in, const int* in_sizes, int n_in,
                              void* d_out, int out_size, void* d_ws, size_t ws_size,
                              hipStream_t stream) {
  const float* feats   = (const float*)d_in[0];
  const int*   indices = (const int*)d_in[1];
  const float* W1 = (const float*)d_in[3];  const float* b1 = (const float*)d_in[4];
  const float* W2 = (const float*)d_in[5];  const float* b2 = (const float*)d_in[6];
  const float* W3 = (const float*)d_in[7];  const float* b3 = (const float*)d_in[8];
  const float* W4 = (const float*)d_in[9];  const float* b4 = (const float*)d_in[10];
  const float* W5 = (const float*)d_in[11]; const float* b5 = (const float*)d_in[12];
  const float* fc1w = (const float*)d_in[13]; const float* fc1b = (const float*)d_in[14];
  const float* fc2w = (const float*)d_in[15]; const float* fc2b = (const float*)d_in[16];
  const float* fc3w = (const float*)d_in[17]; const float* fc3b = (const float*)d_in[18];
  const float* fc4w = (const float*)d_in[19]; const float* fc4b = (const float*)d_in[20];

  const int N = in_sizes[0];            // 8000 points
  const int Npad = ((N + 15) / 16) * 16;

  // carve workspace
  char* base = (char*)d_ws;
  size_t off = 0;
  auto carve = [&](size_t bytes) {
    char* p = base + off;
    off = (off + bytes + 255) & ~(size_t)255;
    return p;
  };
  float* featG = (float*)carve((size_t)2*DG3 * 4);
  int*   idxG  = (int*)  carve((size_t)2*DG3 * 4);
  float* y1    = (float*)carve((size_t)Npad * 32 * 4);
  float* y2    = (float*)carve((size_t)Npad * 40 * 4);
  float* out3  = (float*)carve((size_t)2*S3C*50 * 4);
  float* m3    = (float*)carve((size_t)2*S3C * 4);
  float* x4    = (float*)carve((size_t)2*S4C*50 * 4);
  float* m29   = (float*)carve((size_t)2*S4C * 4);
  float* m14   = (float*)carve((size_t)2*S5C * 4);
  float* m12   = (float*)carve((size_t)2*S6C * 4);
  float* out4  = (float*)carve((size_t)2*S5C*30 * 4);
  float* out5  = (float*)carve((size_t)2*S6C*10 * 4);
  float* flat  = (float*)carve((size_t)2*2160 * 4);
  float* a1    = (float*)carve((size_t)2*500 * 4);
  float* a2    = (float*)carve((size_t)2*200 * 4);
  float* a3    = (float*)carve((size_t)2*50 * 4);

  // zero accumulators / grids (every call: deterministic, graph-safe)
  fillF_k<<<2048, 256, 0, stream>>>(featG, 0.f, (size_t)2*DG3);
  fillI_k<<<2048, 256, 0, stream>>>(idxG, -1, (size_t)2*DG3);
  fillF_k<<<4096, 256, 0, stream>>>(out3, 0.f, (size_t)2*S3C*50);
  fillF_k<<<1024, 256, 0, stream>>>(m3, 0.f, (size_t)2*S3C);

  scatter_k<<<(N + 255) / 256, 256, 0, stream>>>(feats, indices, featG, idxG, N);
  conv1_k<<<N, 32, 0, stream>>>(featG, indices, W1, b1, y1, N);
  conv2_k<<<(N + 15) / 16, 96, 0, stream>>>(y1, idxG, indices, W2, b2, y2, N);
  conv3_k<<<N, 64, 0, stream>>>(y2, idxG, indices, W3, out3, m3, N);
  post3_k<<<(2*S3C + 127) / 128, 128, 0, stream>>>(out3, m3, b3);
  pool4_k<<<(2*S4C*50 + 255) / 256, 256, 0, stream>>>(out3, x4);
  m29_k<<<(2*S4C + 255) / 256, 256, 0, stream>>>(m3, m29);
  m14_k<<<(2*S5C + 255) / 256, 256, 0, stream>>>(m29, m14);
  m12_k<<<(2*S6C + 255) / 256, 256, 0, stream>>>(m14, m12);
  conv4_k<<<(2*S5C*30 + 255) / 256, 256, 0, stream>>>(x4, m14, W4, b4, out4);
  conv5_k<<<(2*S6C*10 + 255) / 256, 256, 0, stream>>>(out4, m12, W5, b5, out5);
  poolflat_k<<<(2*10*216 + 255) / 256, 256, 0, stream>>>(out5, flat);
  fc1_k<<<32, 32, 0, stream>>>(flat, fc1w, fc1b, a1);
  fc2_k<<<(2*200 + 255) / 256, 256, 0, stream>>>(a1, fc2w, fc2b, a2);
  fc3_k<<<(2*50 + 255) / 256, 256, 0, stream>>>(a2, fc3w, fc3b, a3);
  fc4norm_k<<<1, 64, 0, stream>>>(a3, fc4w, fc4b, (float*)d_out);
}